// RecursiveDecoder_90417651516089
// MI455X (gfx1250) — compile-verified
//
#include <hip/hip_runtime.h>
#include <hip/hip_bf16.h>

typedef __attribute__((ext_vector_type(2))) float v2f;
typedef __attribute__((ext_vector_type(8))) float v8f;

// ---------------------------------------------------------------------------
// Generic fp32 WMMA GEMM:  C[M,N] = A[M,K] @ B[K,N] (+ bias[N]) (optional relu)
// One wave per 64x16 C tile: 4 independent 16x16 accumulators share each
// B fragment (V_WMMA_F32_16X16X4_F32, K stepped by 4). 1.5 VMEM per WMMA.
// Requires M%64==0, N%16==0, K%4==0 (true for every call site here).
// ---------------------------------------------------------------------------
__global__ void wmma_gemm_f32(const float* __restrict__ A, int lda,
                              const float* __restrict__ B, int ldb,
                              float* __restrict__ C, int ldc,
                              const float* __restrict__ bias,
                              int M, int N, int K, int do_relu) {
  const int wave = threadIdx.x >> 5;
  const int lane = threadIdx.x & 31;
  const int tilesN = N >> 4;
  const int nTiles = (M >> 6) * tilesN;     // 64-row wave tiles
  const int tile = blockIdx.x * (blockDim.x >> 5) + wave;
  if (tile >= nTiles) return;               // wave-uniform: EXEC stays all-1s
  const int tm = tile / tilesN;             // 64-row block index
  const int tn = tile % tilesN;
  const int half = lane >> 4;               // 0: K pair {0,1}; 1: K pair {2,3}
  const int lmod = lane & 15;               // M index (A) / N index (B)

  const float* A0   = A + (size_t)(tm * 64 + lmod) * lda + 2 * half;
  const float* Bcol = B + (size_t)(2 * half) * ldb + tn * 16 + lmod;
  const size_t lda16 = (size_t)16 * lda;

  v8f acc[4];
#pragma unroll
  for (int t = 0; t < 4; ++t) acc[t] = (v8f){0.f,0.f,0.f,0.f,0.f,0.f,0.f,0.f};

  for (int k = 0; k < K; k += 4) {
    // prefetch the streaming A operand a few k-steps ahead (speculative;
    // out-of-range prefetches are silently dropped)
    __builtin_prefetch(A0 + k + 32, 0, 1);
    v2f b;
    b.x = Bcol[(size_t)k * ldb];
    b.y = Bcol[(size_t)k * ldb + ldb];
    v2f a[4];
#pragma unroll
    for (int t = 0; t < 4; ++t) {
      a[t].x = A0[(size_t)t * lda16 + k];
      a[t].y = A0[(size_t)t * lda16 + k + 1];
    }
#pragma unroll
    for (int t = 0; t < 4; ++t)
      acc[t] = __builtin_amdgcn_wmma_f32_16x16x4_f32(false, a[t], false, b,
                                                     (short)0, acc[t], false, false);
  }

  const int n = tn * 16 + lmod;
  const float bv = bias ? bias[n] : 0.f;
#pragma unroll
  for (int t = 0; t < 4; ++t) {
#pragma unroll
    for (int r = 0; r < 8; ++r) {
      const int m = tm * 64 + t * 16 + r + 8 * half;  // C: VGPR r -> M=r / r+8
      float v = acc[t][r] + bv;
      if (do_relu) v = fmaxf(v, 0.f);
      C[(size_t)m * ldc + n] = v;
    }
  }
}

// ---------------------------------------------------------------------------
// cf = relu(parent_feature @ W_parent + b_parent), written into allf[:, 0:256]
// GEMV (M=1): one thread per output column, pf staged in LDS.
// ---------------------------------------------------------------------------
__global__ void parent_kernel(const float* __restrict__ pf,
                              const float* __restrict__ W,
                              const float* __restrict__ b,
                              float* __restrict__ allf) {
  __shared__ float spf[256];
  spf[threadIdx.x] = pf[threadIdx.x];
  __syncthreads();
  const int o = blockIdx.x * 256 + threadIdx.x;      // 0..32767
  float s = b[o];
  const float* Wc = W + o;
#pragma unroll 4
  for (int f = 0; f < 256; ++f) s = fmaf(spf[f], Wc[(size_t)f * 32768], s);
  allf[(size_t)(o >> 8) * 768 + (o & 255)] = fmaxf(s, 0.f);
}

// ---------------------------------------------------------------------------
// child_exists_logits = cf @ W_exists + b_exists ; exflag = logits > 0
// ---------------------------------------------------------------------------
__global__ void exists_kernel(const float* __restrict__ allf,
                              const float* __restrict__ W,
                              const float* __restrict__ b,
                              float* __restrict__ logits,
                              float* __restrict__ exflag) {
  const int m = threadIdx.x;                // 128 threads, 1 block
  float s = b[0];
  const float* row = allf + (size_t)m * 768;
  for (int h = 0; h < 256; ++h) s = fmaf(row[h], W[h], s);
  logits[m] = s;
  exflag[m] = (s > 0.f) ? 1.f : 0.f;
}

// ---------------------------------------------------------------------------
// el[i,j,:] = relu(Ae[i,:] + Be[j,:])       (b_edge_latent pre-folded into Ae)
// ee[i,j,t] = sum_h el[i,j,h] * W_ee[t,h] + b_ee[t]   (block reduction)
// One block per (i,j) pair; thread = h.
// ---------------------------------------------------------------------------
__global__ void edge_kernel(const float* __restrict__ Ae,
                            const float* __restrict__ Be,
                            float* __restrict__ el,
                            float* __restrict__ ee_ws,
                            float* __restrict__ ee_out,
                            const float* __restrict__ b_ee,
                            const float* __restrict__ W_ee) {
  const int ij = blockIdx.x;
  const int i = ij >> 7, j = ij & 127;
  const int h = threadIdx.x;
  const float v = fmaxf(Ae[i * 256 + h] + Be[j * 256 + h], 0.f);
  el[(size_t)ij * 256 + h] = v;

  __shared__ float s[256];
  for (int t = 0; t < 4; ++t) {
    s[h] = v * W_ee[t * 256 + h];
    __syncthreads();
    for (int off = 128; off > 0; off >>= 1) {
      if (h < off) s[h] += s[h + off];
      __syncthreads();
    }
    if (h == 0) {
      const float e = s[0] + b_ee[t];
      ee_ws[(size_t)ij * 4 + t] = e;
      ee_out[(size_t)ij * 4 + t] = e;
    }
    __syncthreads();
  }
}

// ---------------------------------------------------------------------------
// cf_new[i,h] = max_{j,t} relu(P1[i,h] + P2[j,h] + G[ij,h] + ee[i,j,t]*W4[t,h])
//              * mask(i,j,t)        (b_op pre-folded into P1)
// One block per i, thread = h. ee/exists are wave-uniform (LDS scalars).
// ---------------------------------------------------------------------------
__global__ void op_kernel(const float* __restrict__ P1,
                          const float* __restrict__ P2,
                          const float* __restrict__ G,
                          const float* __restrict__ ee,
                          const float* __restrict__ exflag,
                          const float* __restrict__ W4,
                          float* __restrict__ cf_out) {
  const int i = blockIdx.x;
  const int h = threadIdx.x;
  __shared__ float s_ee[512];
  __shared__ float s_ex[128];
  s_ee[h]       = ee[(size_t)i * 512 + h];
  s_ee[256 + h] = ee[(size_t)i * 512 + 256 + h];
  if (h < 128) s_ex[h] = exflag[h];
  __syncthreads();

  float best = 0.f;   // relu(pre)*mask >= 0 always, so 0 is the exact identity
  if (exflag[i] > 0.f) {
    const float w40 = W4[h], w41 = W4[256 + h], w42 = W4[512 + h], w43 = W4[768 + h];
    const float p1 = P1[(size_t)i * 256 + h];
    const float* Grow = G + (size_t)i * 128 * 256 + h;
    for (int j = 0; j < 128; ++j) {
      if (s_ex[j] == 0.f) continue;               // uniform across the block
      const float base = p1 + P2[(size_t)j * 256 + h] + Grow[(size_t)j * 256];
      const float e0 = s_ee[j * 4 + 0];
      const float e1 = s_ee[j * 4 + 1];
      const float e2 = s_ee[j * 4 + 2];
      const float e3 = s_ee[j * 4 + 3];
      if (e0 > 0.f) best = fmaxf(best, fmaf(e0, w40, base));
      if (e1 > 0.f) best = fmaxf(best, fmaf(e1, w41, base));
      if (e2 > 0.f) best = fmaxf(best, fmaf(e2, w42, base));
      if (e3 > 0.f) best = fmaxf(best, fmaf(e3, w43, base));
    }
  }
  cf_out[(size_t)i * 768 + h] = best;             // into its allf column block
}

// ---------------------------------------------------------------------------
// child_sem_logits = ch @ W_sem + b_sem   (N=57, not WMMA-friendly -> scalar)
// ---------------------------------------------------------------------------
__global__ void sem_kernel(const float* __restrict__ ch,
                           const float* __restrict__ W,
                           const float* __restrict__ b,
                           float* __restrict__ out) {
  const int o = blockIdx.x * blockDim.x + threadIdx.x;
  if (o >= 128 * 57) return;
  const int m = o / 57, t = o % 57;
  float s = b[t];
  const float* row = ch + (size_t)m * 256;
  for (int h = 0; h < 256; ++h) s = fmaf(row[h], W[(size_t)h * 57 + t], s);
  out[o] = s;
}

// ---------------------------------------------------------------------------
extern "C" void kernel_launch(void* const* d_in, const int* in_sizes, int n_in,
                              void* d_out, int out_size, void* d_ws, size_t ws_size,
                              hipStream_t stream) {
  const float* pf       = (const float*)d_in[0];
  const float* W_parent = (const float*)d_in[1];
  const float* b_parent = (const float*)d_in[2];
  const float* W_exists = (const float*)d_in[3];
  const float* b_exists = (const float*)d_in[4];
  const float* W_el     = (const float*)d_in[5];   // (512,256): top=rows 0..255
  const float* b_el     = (const float*)d_in[6];
  const float* W_ee     = (const float*)d_in[7];   // (4,256)
  const float* b_ee     = (const float*)d_in[8];
  const float* W_op     = (const float*)d_in[9];   // (2,772,256)
  const float* b_op     = (const float*)d_in[10];  // (2,256)
  const float* W_child  = (const float*)d_in[11];  // (768,256)
  const float* b_child  = (const float*)d_in[12];
  const float* W_sem    = (const float*)d_in[13];  // (256,57)
  const float* b_sem    = (const float*)d_in[14];
  const float* W_child2 = (const float*)d_in[15];  // (256,256)
  const float* b_child2 = (const float*)d_in[16];

  float* out_feats = (float*)d_out;            // (128,256) = 32768
  float* out_sem   = out_feats + 128 * 256;    // (128,57)  = 7296
  float* out_ex    = out_sem + 128 * 57;       // (128,1)   = 128
  float* out_ee    = out_ex + 128;             // (128,128,4) = 65536

  float* ws     = (float*)d_ws;
  float* allf   = ws;                 // 128*768 = 98304 : [cf0 | cf1 | cf2]
  float* Ae     = allf + 98304;       // 32768
  float* Be     = Ae + 32768;         // 32768
  float* P1     = Be + 32768;         // 32768
  float* P2     = P1 + 32768;         // 32768
  float* exflag = P2 + 32768;         // 128
  float* ee     = exflag + 128;       // 65536
  float* ch     = ee + 65536;         // 32768
  float* el     = ch + 32768;         // 16384*256 = 4194304
  float* G      = el + 4194304;       // 16384*256 = 4194304

  // Stage 0: cf = relu(pf @ W_parent + b_parent) -> allf[:, 0:256]
  parent_kernel<<<128, 256, 0, stream>>>(pf, W_parent, b_parent, allf);
  // Stage 1: exists logits + flags
  exists_kernel<<<1, 128, 0, stream>>>(allf, W_exists, b_exists, out_ex, exflag);
  // Stage 2: factored edge latent: Ae = cf@We_top + b_el ; Be = cf@We_bot
  //   (M=128 -> 2x16 = 32 wave tiles -> 4 blocks of 8 waves)
  wmma_gemm_f32<<<4, 256, 0, stream>>>(allf, 768, W_el, 256, Ae, 256, b_el,
                                       128, 256, 256, 0);
  wmma_gemm_f32<<<4, 256, 0, stream>>>(allf, 768, W_el + 256 * 256, 256, Be, 256,
                                       nullptr, 128, 256, 256, 0);
  // Stage 3: el = relu(Ae[i]+Be[j]) ; ee = el @ W_ee^T + b_ee
  edge_kernel<<<16384, 256, 0, stream>>>(Ae, Be, el, ee, out_ee, b_ee, W_ee);

  // Stage 4: two message-passing iterations
  for (int it = 0; it < 2; ++it) {
    const float* cfc = allf + it * 256;                    // current cf, ld 768
    const float* Wo  = W_op + (size_t)it * 772 * 256;
    wmma_gemm_f32<<<4, 256, 0, stream>>>(cfc, 768, Wo, 256, P1, 256,
                                         b_op + it * 256, 128, 256, 256, 0);
    wmma_gemm_f32<<<4, 256, 0, stream>>>(cfc, 768, Wo + 256 * 256, 256, P2, 256,
                                         nullptr, 128, 256, 256, 0);
    // big one: G = el @ W3  (16384 x 256 x 256) -> 256x16 = 4096 tiles
    wmma_gemm_f32<<<512, 256, 0, stream>>>(el, 256, Wo + 512 * 256, 256, G, 256,
                                           nullptr, 16384, 256, 256, 0);
    op_kernel<<<128, 256, 0, stream>>>(P1, P2, G, ee, exflag, Wo + 768 * 256,
                                       allf + (it + 1) * 256);
  }

  // Stage 5: heads
  wmma_gemm_f32<<<4, 256, 0, stream>>>(allf, 768, W_child, 256, ch, 256, b_child,
                                       128, 768, 256, 1);
  wmma_gemm_f32<<<4, 256, 0, stream>>>(ch, 256, W_child2, 256, out_feats, 256,
                                       b_child2, 128, 256, 256, 1);
  sem_kernel<<<(128 * 57 + 255) / 256, 256, 0, stream>>>(ch, W_sem, b_sem, out_sem);
}